// EdgeConv_85469849190810
// MI455X (gfx1250) — compile-verified
//
#include <hip/hip_runtime.h>
#include <math.h>

#define B_    16
#define N_    2048
#define K_    20
#define CIN   6
#define COUT  64
#define EPS_  1e-5f
#define SLOPE 0.2f

#define NEDGE (B_ * N_ * K_)          // 655360
#define NTILE (NEDGE / 16)            // 40960
#define NOUT  (B_ * N_ * COUT)        // 2097152

typedef float v2f __attribute__((ext_vector_type(2)));
typedef float v8f __attribute__((ext_vector_type(8)));

// ---- order-preserving float <-> uint encoding for atomic max ----
__device__ __forceinline__ unsigned fenc(float f) {
    unsigned u = __float_as_uint(f);
    return (u & 0x80000000u) ? ~u : (u | 0x80000000u);
}
__device__ __forceinline__ float fdec(unsigned u) {
    return __uint_as_float((u & 0x80000000u) ? (u & 0x7FFFFFFFu) : ~u);
}

// ======================= init =======================
__global__ void ec_init_kernel(unsigned* __restrict__ keys, float* __restrict__ stats) {
    int gid = blockIdx.x * blockDim.x + threadIdx.x;
    int stride = gridDim.x * blockDim.x;
    for (int i = gid; i < NOUT; i += stride) keys[i] = 0u;   // encoding-minimum
    if (gid < 2 * COUT) stats[gid] = 0.0f;
}

// ======================= KNN (top-K by d2, stable ties) =======================
__global__ void __launch_bounds__(256) ec_knn_kernel(const float* __restrict__ pos,
                                                     int* __restrict__ idx) {
    const int tid = threadIdx.x;
    const int b   = blockIdx.x >> 3;                 // 8 blocks per cloud
    const int i   = ((blockIdx.x & 7) << 8) + tid;   // query point

    const float* pb = pos + (size_t)b * N_ * 3;
    const float qx = pb[3 * i + 0], qy = pb[3 * i + 1], qz = pb[3 * i + 2];
    const float sqi = qx * qx + qy * qy + qz * qz;

    __shared__ float sx[256], sy[256], sz[256], ssq[256];

    float bd[K_];
    int   bi[K_];
#pragma unroll
    for (int k = 0; k < K_; ++k) { bd[k] = INFINITY; bi[k] = 0; }

    for (int t = 0; t < N_; t += 256) {
        int j = t + tid;
        float x = pb[3 * j + 0], y = pb[3 * j + 1], z = pb[3 * j + 2];
        sx[tid] = x; sy[tid] = y; sz[tid] = z;
        ssq[tid] = x * x + y * y + z * z;
        __syncthreads();
        for (int jj = 0; jj < 256; ++jj) {
            // mirror reference formula: sq_i + sq_j - 2*dot
            float dot = qx * sx[jj] + qy * sy[jj] + qz * sz[jj];
            float d   = sqi + ssq[jj] - 2.0f * dot;
            if (d < bd[K_ - 1]) {                    // strict: earlier index wins ties
                int p = K_ - 1;
                while (p > 0 && bd[p - 1] > d) {
                    bd[p] = bd[p - 1]; bi[p] = bi[p - 1]; --p;
                }
                bd[p] = d; bi[p] = t + jj;
            }
        }
        __syncthreads();
    }
    int base = (b * N_ + i) * K_;
#pragma unroll
    for (int k = 0; k < K_; ++k) idx[base + k] = bi[k];
}

// ======================= shared WMMA tile machinery =======================
// B fragments (W is [6][64] row-major), per wave, uniform across tiles.
__device__ __forceinline__ void ec_load_B(const float* __restrict__ W, int n, bool hi,
                                          v2f b0[4], v2f b1[4]) {
    const int kb = hi ? 2 : 0;
#pragma unroll
    for (int t = 0; t < 4; ++t) {
        int o = t * 16 + n;
        b0[t].x = W[(kb + 0) * COUT + o];            // channels kb, kb+1
        b0[t].y = W[(kb + 1) * COUT + o];
        b1[t].x = hi ? 0.0f : W[4 * COUT + o];       // channels 4,5 (6,7 are zero-pad)
        b1[t].y = hi ? 0.0f : W[5 * COUT + o];
    }
}

// Computes h-tile (pre-bias) for 16 edges x 64 channels into acc[4] (v8f each),
// returns this lane's edge segment id (b*N + j) for edge m = lane&15.
__device__ __forceinline__ int ec_tile(const float* __restrict__ pos,
                                       const int* __restrict__ idx,
                                       int tile, int lane, bool hi,
                                       const v2f b0[4], const v2f b1[4],
                                       v8f acc[4]) {
    const int m = lane & 15;
    const int e = tile * 16 + m;
    const int b = e / (N_ * K_);
    const int r = e % (N_ * K_);
    const int i = r / K_;
    const int j = idx[e];
    const float* pb = pos + (size_t)b * N_ * 3;
    const float px = pb[3 * j + 0], py = pb[3 * j + 1], pz = pb[3 * j + 2];
    const float qx = pb[3 * i + 0], qy = pb[3 * i + 1], qz = pb[3 * i + 2];
    // channels: 0:px 1:py 2:pz 3:qx-px 4:qy-py 5:qz-pz
    v2f a0, a1;
    if (!hi) { a0.x = px; a0.y = py;      a1.x = qy - py; a1.y = qz - pz; }
    else     { a0.x = pz; a0.y = qx - px; a1.x = 0.0f;    a1.y = 0.0f;    }

#pragma unroll
    for (int t = 0; t < 4; ++t) {
#pragma unroll
        for (int rr = 0; rr < 8; ++rr) acc[t][rr] = 0.0f;
        acc[t] = __builtin_amdgcn_wmma_f32_16x16x4_f32(false, a0, false, b0[t],
                                                       (short)0, acc[t], false, false);
        acc[t] = __builtin_amdgcn_wmma_f32_16x16x4_f32(false, a1, false, b1[t],
                                                       (short)0, acc[t], false, false);
    }
    return b * N_ + j;
}

// ======================= pass 1: BN statistics =======================
__global__ void __launch_bounds__(256) ec_stats_kernel(const float* __restrict__ pos,
                                                       const int* __restrict__ idx,
                                                       const float* __restrict__ W,
                                                       const float* __restrict__ bias,
                                                       float* __restrict__ stats) {
    __shared__ float ssum[COUT], ssq[COUT];
    if (threadIdx.x < COUT) { ssum[threadIdx.x] = 0.0f; ssq[threadIdx.x] = 0.0f; }
    __syncthreads();

    const int lane = threadIdx.x & 31;
    const bool hi  = lane >= 16;
    const int n    = lane & 15;
    const int wid  = (blockIdx.x * blockDim.x + threadIdx.x) >> 5;
    const int nw   = (gridDim.x * blockDim.x) >> 5;

    v2f b0[4], b1[4];
    ec_load_B(W, n, hi, b0, b1);
    float bs[4], psum[4], psq[4];
#pragma unroll
    for (int t = 0; t < 4; ++t) { bs[t] = bias[t * 16 + n]; psum[t] = 0.0f; psq[t] = 0.0f; }

    for (int tile = wid; tile < NTILE; tile += nw) {
        v8f acc[4];
        (void)ec_tile(pos, idx, tile, lane, hi, b0, b1, acc);
#pragma unroll
        for (int t = 0; t < 4; ++t)
#pragma unroll
            for (int r = 0; r < 8; ++r) {
                float v = acc[t][r] + bs[t];
                psum[t] += v;
                psq[t]  += v * v;
            }
    }
#pragma unroll
    for (int t = 0; t < 4; ++t) {
        psum[t] += __shfl_xor(psum[t], 16, 32);      // lanes L/L+16 share channel
        psq[t]  += __shfl_xor(psq[t], 16, 32);
    }
    if (!hi) {
#pragma unroll
        for (int t = 0; t < 4; ++t) {
            atomicAdd(&ssum[t * 16 + n], psum[t]);   // ds_add_f32
            atomicAdd(&ssq[t * 16 + n], psq[t]);
        }
    }
    __syncthreads();
    if (threadIdx.x < COUT) {
        atomicAdd(&stats[threadIdx.x], ssum[threadIdx.x]);
        atomicAdd(&stats[COUT + threadIdx.x], ssq[threadIdx.x]);
    }
}

// ======================= BN scale/shift =======================
__global__ void ec_finalize_kernel(const float* __restrict__ gamma,
                                   const float* __restrict__ beta,
                                   const float* __restrict__ stats,
                                   float* __restrict__ scsh) {
    int c = threadIdx.x;
    if (c < COUT) {
        const float cnt  = (float)NEDGE;
        float mean = stats[c] / cnt;
        float var  = stats[COUT + c] / cnt - mean * mean;  // biased var (torch BN fwd)
        float s    = rsqrtf(var + EPS_) * gamma[c];
        scsh[c]        = s;
        scsh[COUT + c] = beta[c] - mean * s;
    }
}

// ======================= pass 2: BN + LeakyReLU + segment max =======================
__global__ void __launch_bounds__(256) ec_apply_kernel(const float* __restrict__ pos,
                                                       const int* __restrict__ idx,
                                                       const float* __restrict__ W,
                                                       const float* __restrict__ bias,
                                                       const float* __restrict__ scsh,
                                                       unsigned* __restrict__ keys) {
    const int lane = threadIdx.x & 31;
    const bool hi  = lane >= 16;
    const int n    = lane & 15;
    const int wid  = (blockIdx.x * blockDim.x + threadIdx.x) >> 5;
    const int nw   = (gridDim.x * blockDim.x) >> 5;

    v2f b0[4], b1[4];
    ec_load_B(W, n, hi, b0, b1);
    float bs[4], sc[4], sh[4];
#pragma unroll
    for (int t = 0; t < 4; ++t) {
        int c = t * 16 + n;
        bs[t] = bias[c];
        sc[t] = scsh[c];
        sh[t] = scsh[COUT + c];
    }

    for (int tile = wid; tile < NTILE; tile += nw) {
        v8f acc[4];
        int seg_own = ec_tile(pos, idx, tile, lane, hi, b0, b1, acc);
#pragma unroll
        for (int t = 0; t < 4; ++t) {
            int c = t * 16 + n;
#pragma unroll
            for (int r = 0; r < 8; ++r) {
                int M   = hi ? (r + 8) : r;          // C-layout row this VGPR holds
                int seg = __shfl(seg_own, M, 32);    // lane M owns edge M's segment
                float v = acc[t][r] + bs[t];
                v = v * sc[t] + sh[t];
                v = (v > 0.0f) ? v : SLOPE * v;
                atomicMax(&keys[(size_t)seg * COUT + c], fenc(v));
            }
        }
    }
}

// ======================= decode =======================
__global__ void ec_decode_kernel(unsigned* __restrict__ keys, float* __restrict__ outf) {
    int i = blockIdx.x * blockDim.x + threadIdx.x;
    if (i < NOUT) {
        unsigned u = keys[i];
        outf[i] = fdec(u);
    }
}

// ======================= launch =======================
extern "C" void kernel_launch(void* const* d_in, const int* in_sizes, int n_in,
                              void* d_out, int out_size, void* d_ws, size_t ws_size,
                              hipStream_t stream) {
    const float* pos   = (const float*)d_in[0];
    const float* W     = (const float*)d_in[1];
    const float* bias  = (const float*)d_in[2];
    const float* gamma = (const float*)d_in[3];
    const float* beta  = (const float*)d_in[4];

    int*   idxbuf = (int*)d_ws;                                      // 655360 ints
    float* stats  = (float*)((char*)d_ws + (size_t)NEDGE * sizeof(int)); // 128 f
    float* scsh   = stats + 2 * COUT;                                // 128 f

    unsigned* keys = (unsigned*)d_out;
    float*    outf = (float*)d_out;

    ec_init_kernel<<<1024, 256, 0, stream>>>(keys, stats);
    ec_knn_kernel<<<(B_ * N_) / 256, 256, 0, stream>>>(pos, idxbuf);
    ec_stats_kernel<<<256, 256, 0, stream>>>(pos, idxbuf, W, bias, stats);
    ec_finalize_kernel<<<1, 64, 0, stream>>>(gamma, beta, stats, scsh);
    ec_apply_kernel<<<256, 256, 0, stream>>>(pos, idxbuf, W, bias, scsh, keys);
    ec_decode_kernel<<<(NOUT + 255) / 256, 256, 0, stream>>>(keys, outf);
}